// DomainAdaptationLoss_81063212744786
// MI455X (gfx1250) — compile-verified
//
#include <hip/hip_runtime.h>

// ---------------- problem constants (fixed by reference) ----------------
#define BATCH 16
#define BC    16            // B*C
#define H     1024
#define W     1024
#define NPIX  (H*W)
#define BINS  64
#define EPSF  1e-6f

#define TILE  32
#define HALO  2
#define HT    (TILE + 2*HALO)   // 36
#define TPB   256
#define TILES_PER_IMG ((H/TILE)*(W/TILE))   // 1024
#define NBLK_FUSED (BC*TILES_PER_IMG)       // 16384
#define MM_BLOCKS 64

// ---------------- CDNA5 feature probes ----------------
#if defined(__AMDGCN__) && __has_builtin(__builtin_amdgcn_global_load_async_to_lds_b32)
#define HAVE_ASYNC_LDS 1
#else
#define HAVE_ASYNC_LDS 0
#endif

#if defined(__AMDGCN__) && __has_builtin(__builtin_amdgcn_wmma_f32_16x16x4_f32)
#define HAVE_WMMA4 1
#else
#define HAVE_WMMA4 0
#endif

typedef __attribute__((ext_vector_type(2))) float v2f;
typedef __attribute__((ext_vector_type(8))) float v8f;

#if HAVE_ASYNC_LDS
// Builtin signature (from hipcc diagnostic): param0 = addrspace(1) int*  ("__device__ int*"),
// param1 = addrspace(3) int*, then two integer immediates (offset, cpol).
typedef __attribute__((address_space(1))) int gint_t;
typedef __attribute__((address_space(3))) int lint_t;
#endif

// ---------------- helpers ----------------
__device__ __forceinline__ float nan2num(float x) {
  if (__builtin_isnan(x)) return 0.f;
  if (__builtin_isinf(x)) return x > 0.f ? 1.f : -1.f;
  return x;
}
__device__ __forceinline__ float clip10(float x) { return fminf(fmaxf(x, 0.f), 10.f); }

// 256-thread block sum; result valid at tid 0. wave32 shfl + 8-slot LDS.
__device__ __forceinline__ float block_sum_256(float v, float* red) {
  #pragma unroll
  for (int o = 16; o > 0; o >>= 1) v += __shfl_down(v, o);
  int lane = threadIdx.x & 31, wid = threadIdx.x >> 5;
  if (lane == 0) red[wid] = v;
  __syncthreads();
  float r = (threadIdx.x < 8) ? red[threadIdx.x] : 0.f;
  if (wid == 0) {
    #pragma unroll
    for (int o = 4; o > 0; o >>= 1) r += __shfl_down(r, o);
  }
  __syncthreads();
  return r;
}

// Sum 64 strided floats with one full wave. On CDNA5 use one
// V_WMMA_F32_16X16X4_F32: A = data (16x4, 2 VGPRs/lane), B = ones(4x16),
// D[m][n] = rowsum(m); per-lane sum of the 8 D regs covers half the rows,
// lane0 + lane16 gives the grand total.
__device__ __forceinline__ float wave_sum64(const float* __restrict__ buf, int stride) {
  int lane = threadIdx.x & 31;
#if HAVE_WMMA4
  v2f a; a.x = buf[(2*lane)*stride]; a.y = buf[(2*lane+1)*stride];
  v2f b; b.x = 1.f; b.y = 1.f;
  v8f c = {0.f,0.f,0.f,0.f,0.f,0.f,0.f,0.f};
  v8f d = __builtin_amdgcn_wmma_f32_16x16x4_f32(false, a, false, b, (short)0, c, false, false);
  float s = d[0]+d[1]+d[2]+d[3]+d[4]+d[5]+d[6]+d[7];
  return __shfl(s, 0) + __shfl(s, 16);
#else
  float s = buf[(2*lane)*stride] + buf[(2*lane+1)*stride];
  #pragma unroll
  for (int o = 16; o > 0; o >>= 1) s += __shfl_down(s, o);
  return __shfl(s, 0);
#endif
}

// ---------------- kernel 0: zero the histogram accumulators ----------------
__global__ void k_zero(unsigned* __restrict__ histG) {
  int i = blockIdx.x * blockDim.x + threadIdx.x;
  if (i < BC * 2 * BINS) histG[i] = 0u;
}

// ---------------- kernel 1: per-channel min/max (pass 1) ----------------
__global__ __launch_bounds__(TPB) void k_minmax(const float* __restrict__ pred,
                                                const float* __restrict__ targ,
                                                float* __restrict__ mmPart) {
  __shared__ float sh[8][4];
  const int ch  = blockIdx.x / MM_BLOCKS;
  const int blk = blockIdx.x % MM_BLOCKS;
  const int seg = NPIX / MM_BLOCKS;                 // 16384
  const size_t base = (size_t)ch * NPIX + (size_t)blk * seg;
  float mnP = 3.4e38f, mxP = -3.4e38f, mnT = 3.4e38f, mxT = -3.4e38f;
  for (int i = threadIdx.x; i < seg; i += TPB) {
    float p = nan2num(pred[base + i]);
    float t = nan2num(targ[base + i]);
    mnP = fminf(mnP, p); mxP = fmaxf(mxP, p);
    mnT = fminf(mnT, t); mxT = fmaxf(mxT, t);
  }
  #pragma unroll
  for (int o = 16; o > 0; o >>= 1) {
    mnP = fminf(mnP, __shfl_down(mnP, o)); mxP = fmaxf(mxP, __shfl_down(mxP, o));
    mnT = fminf(mnT, __shfl_down(mnT, o)); mxT = fmaxf(mxT, __shfl_down(mxT, o));
  }
  int lane = threadIdx.x & 31, wid = threadIdx.x >> 5;
  if (lane == 0) { sh[wid][0] = mnP; sh[wid][1] = mxP; sh[wid][2] = mnT; sh[wid][3] = mxT; }
  __syncthreads();
  if (threadIdx.x == 0) {
    for (int w2 = 1; w2 < 8; w2++) {
      mnP = fminf(mnP, sh[w2][0]); mxP = fmaxf(mxP, sh[w2][1]);
      mnT = fminf(mnT, sh[w2][2]); mxT = fmaxf(mxT, sh[w2][3]);
    }
    float* o = mmPart + (size_t)blockIdx.x * 4;
    o[0] = mnP; o[1] = mxP; o[2] = mnT; o[3] = mxT;
  }
}

// ---------------- kernel 2: fold min/max -> normalization params ----------------
__global__ void k_minmax_reduce(const float* __restrict__ mmPart, float* __restrict__ params) {
  const int ch = blockIdx.x;
  const int t  = threadIdx.x;                        // 0..63
  const float* p = mmPart + (size_t)(ch * MM_BLOCKS + t) * 4;
  float mnP = p[0], mxP = p[1], mnT = p[2], mxT = p[3];
  #pragma unroll
  for (int o = 16; o > 0; o >>= 1) {
    mnP = fminf(mnP, __shfl_down(mnP, o)); mxP = fmaxf(mxP, __shfl_down(mxP, o));
    mnT = fminf(mnT, __shfl_down(mnT, o)); mxT = fmaxf(mxT, __shfl_down(mxT, o));
  }
  __shared__ float sh[2][4];
  int wid = t >> 5, lane = t & 31;
  if (lane == 0) { sh[wid][0] = mnP; sh[wid][1] = mxP; sh[wid][2] = mnT; sh[wid][3] = mxT; }
  __syncthreads();
  if (t == 0) {
    mnP = fminf(sh[0][0], sh[1][0]); mxP = fmaxf(sh[0][1], sh[1][1]);
    mnT = fminf(sh[0][2], sh[1][2]); mxT = fmaxf(sh[0][3], sh[1][3]);
    float dP = mxP - mnP; bool cP = dP < 1e-6f;
    float dT = mxT - mnT; bool cT = dT < 1e-6f;
    float* pr = params + ch * 8;
    pr[0] = mnP; pr[1] = cP ? 0.f : 1.f / dP; pr[2] = cP ? 1.f : 0.f;
    pr[3] = mnT; pr[4] = cT ? 0.f : 1.f / dT; pr[5] = cT ? 1.f : 0.f;
  }
}

// ---------------- kernel 3: fused tiled pass (moments+hist+sobel+local-std) ----
__global__ __launch_bounds__(TPB) void k_fused(const float* __restrict__ pred,
                                               const float* __restrict__ targ,
                                               const float* __restrict__ params,
                                               float* __restrict__ momPart,
                                               float* __restrict__ gtPart,
                                               unsigned* __restrict__ histG) {
  __shared__ float sP[HT * HT];        // 1296
  __shared__ float sT[HT * HT];
  __shared__ float rsP[HT * TILE];     // 1152: horizontal 5-tap sums
  __shared__ float rs2P[HT * TILE];
  __shared__ float rsT[HT * TILE];
  __shared__ float rs2T[HT * TILE];
  __shared__ unsigned histL[2 * BINS];
  __shared__ float red[8];

  const int tid    = threadIdx.x;
  const int ch     = blockIdx.x / TILES_PER_IMG;
  const int tileId = blockIdx.x % TILES_PER_IMG;
  const int y0 = (tileId >> 5) * TILE, x0 = (tileId & 31) * TILE;
  const float* gP = pred + (size_t)ch * NPIX;
  const float* gT = targ + (size_t)ch * NPIX;

  const float minP = params[ch*8+0], invP = params[ch*8+1], cstP = params[ch*8+2];
  const float minT = params[ch*8+3], invT = params[ch*8+4], cstT = params[ch*8+5];

  if (tid < 2 * BINS) histL[tid] = 0u;

  // Stage raw halo tiles into LDS. CDNA5: async DMA global->LDS (ASYNCcnt).
  for (int i = tid; i < HT * HT; i += TPB) {
    int hy = i / HT, hx = i - hy * HT;
    int gy = y0 + hy - HALO, gx = x0 + hx - HALO;
    int gyc = min(max(gy, 0), H - 1), gxc = min(max(gx, 0), W - 1);
    size_t off = (size_t)gyc * W + gxc;
#if HAVE_ASYNC_LDS
    __builtin_amdgcn_global_load_async_to_lds_b32((gint_t*)(gP + off), (lint_t*)&sP[i], 0, 0);
    __builtin_amdgcn_global_load_async_to_lds_b32((gint_t*)(gT + off), (lint_t*)&sT[i], 0, 0);
#else
    sP[i] = gP[off];
    sT[i] = gT[off];
#endif
  }
#if HAVE_ASYNC_LDS
#if __has_builtin(__builtin_amdgcn_s_wait_asynccnt)
  __builtin_amdgcn_s_wait_asynccnt(0);
#else
  asm volatile("s_wait_asynccnt 0" ::: "memory");
#endif
#endif
  __syncthreads();

  // nan_to_num + per-channel min-max normalize in place; OOB halo -> 0 (zero pad).
  for (int i = tid; i < HT * HT; i += TPB) {
    int hy = i / HT, hx = i - hy * HT;
    int gy = y0 + hy - HALO, gx = x0 + hx - HALO;
    bool inb = (gy >= 0) & (gy < H) & (gx >= 0) & (gx < W);
    float pv = 0.f, tv = 0.f;
    if (inb) {
      float rp = nan2num(sP[i]);
      float rt = nan2num(sT[i]);
      pv = (cstP != 0.f) ? 0.5f : fminf(fmaxf((rp - minP) * invP, 0.f), 1.f);
      tv = (cstT != 0.f) ? 0.5f : fminf(fmaxf((rt - minT) * invT, 0.f), 1.f);
    }
    sP[i] = pv; sT[i] = tv;
  }
  __syncthreads();

  // Separable 5x5 box: horizontal running sums (value and square).
  for (int i = tid; i < HT * TILE; i += TPB) {
    int r = i / TILE, c = i - r * TILE;
    int b = r * HT + c + HALO;
    float s = 0.f, s2 = 0.f, u;
    #pragma unroll
    for (int d = -2; d <= 2; d++) { u = sP[b + d]; s += u; s2 += u * u; }
    rsP[i] = s; rs2P[i] = s2;
    s = 0.f; s2 = 0.f;
    #pragma unroll
    for (int d = -2; d <= 2; d++) { u = sT[b + d]; s += u; s2 += u * u; }
    rsT[i] = s; rs2T[i] = s2;
  }
  __syncthreads();

  float sp = 0, st = 0, spp = 0, stt = 0, spt = 0, gsum = 0, tsum = 0;
  #pragma unroll
  for (int k = 0; k < 4; k++) {
    int pi = tid + k * TPB;
    int y = pi >> 5, x = pi & 31;
    int c0 = (y + HALO) * HT + (x + HALO);
    float pv = sP[c0], tv = sT[c0];
    sp += pv; st += tv; spp += pv * pv; stt += tv * tv; spt += pv * tv;
    int bp = (int)(pv * 64.f); bp = bp > 63 ? 63 : (bp < 0 ? 0 : bp);
    int bt = (int)(tv * 64.f); bt = bt > 63 ? 63 : (bt < 0 ? 0 : bt);
    atomicAdd(&histL[bp], 1u);
    atomicAdd(&histL[64 + bt], 1u);
    { // Sobel magnitude diff
      float a00 = sP[c0-HT-1], a01 = sP[c0-HT], a02 = sP[c0-HT+1];
      float a10 = sP[c0-1],                     a12 = sP[c0+1];
      float a20 = sP[c0+HT-1], a21 = sP[c0+HT], a22 = sP[c0+HT+1];
      float gxp = (a02 + 2.f*a12 + a22) - (a00 + 2.f*a10 + a20);
      float gyp = (a20 + 2.f*a21 + a22) - (a00 + 2.f*a01 + a02);
      float b00 = sT[c0-HT-1], b01 = sT[c0-HT], b02 = sT[c0-HT+1];
      float b10 = sT[c0-1],                     b12 = sT[c0+1];
      float b20 = sT[c0+HT-1], b21 = sT[c0+HT], b22 = sT[c0+HT+1];
      float gxt = (b02 + 2.f*b12 + b22) - (b00 + 2.f*b10 + b20);
      float gyt = (b20 + 2.f*b21 + b22) - (b00 + 2.f*b01 + b02);
      gsum += fabsf(sqrtf(gxp*gxp + gyp*gyp + EPSF) - sqrtf(gxt*gxt + gyt*gyt + EPSF));
    }
    { // local std diff (vertical 5-tap over row sums)
      float s = 0.f, s2 = 0.f;
      #pragma unroll
      for (int d = 0; d < 5; d++) { s += rsP[(y+d)*TILE + x]; s2 += rs2P[(y+d)*TILE + x]; }
      float mu  = s * (1.f/25.f);
      float lsp = sqrtf(fmaxf(s2 * (1.f/25.f) - mu*mu, 0.f) + EPSF);
      s = 0.f; s2 = 0.f;
      #pragma unroll
      for (int d = 0; d < 5; d++) { s += rsT[(y+d)*TILE + x]; s2 += rs2T[(y+d)*TILE + x]; }
      mu = s * (1.f/25.f);
      float lst = sqrtf(fmaxf(s2 * (1.f/25.f) - mu*mu, 0.f) + EPSF);
      tsum += fabsf(lsp - lst);
    }
  }
  __syncthreads();

  float r;
  r = block_sum_256(sp,  red); if (tid == 0) momPart[(size_t)blockIdx.x*5+0] = r;
  r = block_sum_256(st,  red); if (tid == 0) momPart[(size_t)blockIdx.x*5+1] = r;
  r = block_sum_256(spp, red); if (tid == 0) momPart[(size_t)blockIdx.x*5+2] = r;
  r = block_sum_256(stt, red); if (tid == 0) momPart[(size_t)blockIdx.x*5+3] = r;
  r = block_sum_256(spt, red); if (tid == 0) momPart[(size_t)blockIdx.x*5+4] = r;
  r = block_sum_256(gsum, red); if (tid == 0) gtPart[(size_t)blockIdx.x*2+0] = r;
  r = block_sum_256(tsum, red); if (tid == 0) gtPart[(size_t)blockIdx.x*2+1] = r;

  if (tid < 2 * BINS) atomicAdd(&histG[ch * 2 * BINS + tid], histL[tid]);  // integer: deterministic
}

// ---------------- kernel 4: fold moment partials per channel ----------------
__global__ __launch_bounds__(TPB) void k_moment_reduce(const float* __restrict__ momPart,
                                                       float* __restrict__ chanMom) {
  __shared__ float red[8];
  const int ch = blockIdx.x;
  float a0 = 0, a1 = 0, a2 = 0, a3 = 0, a4 = 0;
  for (int blk = threadIdx.x; blk < TILES_PER_IMG; blk += TPB) {
    const float* m = momPart + ((size_t)ch * TILES_PER_IMG + blk) * 5;
    a0 += m[0]; a1 += m[1]; a2 += m[2]; a3 += m[3]; a4 += m[4];
  }
  float r;
  r = block_sum_256(a0, red); if (threadIdx.x == 0) chanMom[ch*5+0] = r;
  r = block_sum_256(a1, red); if (threadIdx.x == 0) chanMom[ch*5+1] = r;
  r = block_sum_256(a2, red); if (threadIdx.x == 0) chanMom[ch*5+2] = r;
  r = block_sum_256(a3, red); if (threadIdx.x == 0) chanMom[ch*5+3] = r;
  r = block_sum_256(a4, red); if (threadIdx.x == 0) chanMom[ch*5+4] = r;
}

// ---------------- kernel 5: fold grad/texture partials 16384 -> 64 ----------
__global__ __launch_bounds__(TPB) void k_gt_reduce(const float* __restrict__ gtPart,
                                                   float* __restrict__ gt2) {
  __shared__ float red[8];
  const size_t row = (size_t)blockIdx.x * TPB + threadIdx.x;
  float g = gtPart[row*2+0], t = gtPart[row*2+1];
  float r;
  r = block_sum_256(g, red); if (threadIdx.x == 0) gt2[blockIdx.x*2+0] = r;
  r = block_sum_256(t, red); if (threadIdx.x == 0) gt2[blockIdx.x*2+1] = r;
}

// ---------------- kernel 6: final combine (one wave, WMMA reductions) -------
__global__ void k_final(const float* __restrict__ chanMom, const float* __restrict__ gt2,
                        const unsigned* __restrict__ histG, float* __restrict__ out) {
  const int lane = threadIdx.x;   // 0..31, one wave32
  float contrast = 0.f, structure = 0.f, nccl = 0.f;
  if (lane < BC) {
    const float* m = chanMom + lane * 5;
    const float n = (float)NPIX;
    float mu_p = m[0] / n, mu_t = m[1] / n;
    float var_p = fmaxf(m[2] / n - mu_p * mu_p, 0.f);
    float var_t = fmaxf(m[3] / n - mu_t * mu_t, 0.f);
    float cov   = m[4] / n - mu_p * mu_t;
    contrast = fabsf(sqrtf(var_p + EPSF) - sqrtf(var_t + EPSF));
    const float c1 = 1e-4f, c2 = 9e-4f;
    float ssim = ((2.f*mu_p*mu_t + c1) * (2.f*cov + c2)) /
                 ((mu_p*mu_p + mu_t*mu_t + c1) * (var_p + var_t + c2));
    structure = 1.f - ssim;
    nccl = 1.f - cov / (sqrtf(var_p * var_t) + EPSF);
  }
  #pragma unroll
  for (int o = 16; o > 0; o >>= 1) {
    contrast  += __shfl_down(contrast, o);
    structure += __shfl_down(structure, o);
    nccl      += __shfl_down(nccl, o);
  }
  contrast  = __shfl(contrast, 0)  * (1.f / BC);
  structure = __shfl(structure, 0) * (1.f / BC);
  nccl      = __shfl(nccl, 0)      * (1.f / BC);

  float hacc = 0.f;
  for (int i = lane; i < BC * BINS; i += 32) {
    int c = i / BINS, b = i - c * BINS;
    float hp = (float)histG[c * 2 * BINS + b]        * (1.f / (float)NPIX);
    float ht = (float)histG[c * 2 * BINS + BINS + b] * (1.f / (float)NPIX);
    hacc += fabsf(hp - ht);
  }
  #pragma unroll
  for (int o = 16; o > 0; o >>= 1) hacc += __shfl_down(hacc, o);
  float hist_loss = __shfl(hacc, 0) * (1.f / BC);

  float gtot = wave_sum64(gt2 + 0, 2);     // v_wmma_f32_16x16x4_f32 reduction
  float ttot = wave_sum64(gt2 + 1, 2);
  float grad_loss    = gtot * (1.f / ((float)BC * (float)NPIX));
  float texture_loss = ttot * (1.f / ((float)BC * (float)NPIX));

  if (lane == 0) {
    float total = clip10(hist_loss) + clip10(contrast) + clip10(structure) +
                  clip10(grad_loss) + clip10(nccl) + clip10(texture_loss);
    out[0] = fminf(fmaxf(total, 0.f), 100.f);
  }
}

// ---------------- host launcher ----------------
extern "C" void kernel_launch(void* const* d_in, const int* in_sizes, int n_in,
                              void* d_out, int out_size, void* d_ws, size_t ws_size,
                              hipStream_t stream) {
  (void)in_sizes; (void)n_in; (void)out_size; (void)ws_size;
  const float* pred = (const float*)d_in[0];
  const float* targ = (const float*)d_in[1];
  float* out = (float*)d_out;

  float* wsf      = (float*)d_ws;
  float* mmPart   = wsf;               // 16*64*4      = 4096 floats
  float* params   = wsf + 4096;        // 16*8         = 128
  float* momPart  = wsf + 4224;        // 16384*5      = 81920
  float* gtPart   = wsf + 86144;       // 16384*2      = 32768
  float* gt2      = wsf + 118912;      // 64*2         = 128
  float* chanMom  = wsf + 119040;      // 16*5         = 80
  unsigned* histG = (unsigned*)(wsf + 119168); // 16*2*64 = 2048 uints

  k_zero<<<8, 256, 0, stream>>>(histG);
  k_minmax<<<BC * MM_BLOCKS, TPB, 0, stream>>>(pred, targ, mmPart);
  k_minmax_reduce<<<BC, 64, 0, stream>>>(mmPart, params);
  k_fused<<<NBLK_FUSED, TPB, 0, stream>>>(pred, targ, params, momPart, gtPart, histG);
  k_moment_reduce<<<BC, TPB, 0, stream>>>(momPart, chanMom);
  k_gt_reduce<<<64, TPB, 0, stream>>>(gtPart, gt2);
  k_final<<<1, 32, 0, stream>>>(chanMom, gt2, histG, out);
}